// TrajEncoder_79147657330867
// MI455X (gfx1250) — compile-verified
//
#include <hip/hip_runtime.h>
#include <cstdint>

// ---------------------------------------------------------------------------
// MI455X (gfx1250) fused implementation.
// FP32 matrix math via V_WMMA_F32_16X16X4_F32 (wave32, 16x16 tile, K-step 4).
// A-frag: lane L holds X[M=L&15, K = 2*(L>>4) + {0,1}]  -> v2f
// B-frag: lane L holds W^T[K = 2*(L>>4) + {0,1}, N=L&15] -> v2f
// C/D   : VGPR r, lanes 0-15 -> (M=r, N=lane), lanes 16-31 -> (M=r+8, N=lane-16)
// ---------------------------------------------------------------------------

typedef __attribute__((ext_vector_type(2))) float v2f;
typedef __attribute__((ext_vector_type(8))) float v8f;

#define DEV __device__ __forceinline__

DEV v8f wmma_f32(v2f a, v2f b, v8f c) {
  // (neg_a, A, neg_b, B, c_mod, C, reuse_a, reuse_b)
  return __builtin_amdgcn_wmma_f32_16x16x4_f32(false, a, false, b, (short)0, c,
                                               false, false);
}

DEV unsigned fkey(float f) {  // monotonic float->uint mapping
  unsigned u = __float_as_uint(f);
  return (u & 0x80000000u) ? ~u : (u | 0x80000000u);
}

DEV unsigned long long shfl_xor_u64(unsigned long long v, int m) {
  int lo = __shfl_xor((int)(unsigned)v, m, 32);
  int hi = __shfl_xor((int)(unsigned)(v >> 32), m, 32);
  return ((unsigned long long)(unsigned)hi << 32) | (unsigned)lo;
}

// ----- problem constants ----------------------------------------------------
#define OBS 17
#define ACT 6
#define HOR 32
#define CTX 64
#define EMBT 5
#define HIDW 512
#define NEMB 126976
#define BATCH 1024
#define TRANSD 24
#define HS 516  // LDS stride for 512-wide hidden activations (bank-friendly)

// output layout (floats) in d_out
#define OFF_CON 0
#define OFF_LOSS 65536
#define OFF_OBS 65537
#define OFF_ACT 655361
#define OFF_TRANSI 845825
#define OFF_ENCX 1435649
#define OFF_NEMB 1501185
#define NEMB_ELEMS 8126464u  // 126976*64

// ---------------------------------------------------------------------------
// Generic 16-row x 512-col layer: each wave computes a 16x64 slice.
// ---------------------------------------------------------------------------
template <int K, int SXs, bool G>
DEV void layer_compute(const float* __restrict__ W, const float* src, int half,
                       int l15, int n0, v8f acc[4]) {
  constexpr int KR = (K + 3) & ~3;
  for (int k = 0; k < KR; k += 4) {
    const int ka = k + 2 * half;
    v2f a = *(const v2f*)(src + l15 * SXs + ka);
#pragma unroll
    for (int t = 0; t < 4; ++t) {
      const int n = n0 + t * 16 + l15;
      const float* wr = W + (size_t)n * K;
      v2f b;
      if (G) {  // K not multiple of 4: clamp (A is zero-padded there)
        const int i0 = ka < K ? ka : K - 1;
        const int i1 = ka + 1 < K ? ka + 1 : K - 1;
        b.x = wr[i0];
        b.y = wr[i1];
      } else {
        b = *(const v2f*)(wr + ka);
      }
      acc[t] = wmma_f32(a, b, acc[t]);
    }
  }
}

DEV void layer_store(float* dst, const float* __restrict__ bias,
                     const v8f acc[4], int half, int l15, int n0, bool relu) {
#pragma unroll
  for (int t = 0; t < 4; ++t) {
    const int n = n0 + t * 16 + l15;
    const float bv = bias[n];
#pragma unroll
    for (int r = 0; r < 8; ++r) {
      const int m = r + 8 * half;
      float v = acc[t][r] + bv;
      if (relu) v = v > 0.f ? v : 0.f;
      dst[m * HS + n] = v;
    }
  }
}

template <int NOUT>
DEV void out_layer(const float* __restrict__ W, const float* __restrict__ bias,
                   const float* src, float* __restrict__ outp, int rowbase,
                   int wave, int half, int l15) {
  constexpr int T = (NOUT + 15) / 16;
  if (wave >= T) return;
  const int n = wave * 16 + l15;
  const int nc = n < NOUT ? n : NOUT - 1;  // clamp: garbage cols discarded
  const float* wr = W + (size_t)nc * HIDW;
  v8f acc;
#pragma unroll
  for (int r = 0; r < 8; ++r) acc[r] = 0.f;
  for (int k = 0; k < HIDW; k += 4) {
    const int ka = k + 2 * half;
    v2f a = *(const v2f*)(src + l15 * HS + ka);
    v2f b = *(const v2f*)(wr + ka);
    acc = wmma_f32(a, b, acc);
  }
  if (n < NOUT) {
    const float bv = bias[n];
#pragma unroll
    for (int r = 0; r < 8; ++r) {
      const int m = r + 8 * half;
      outp[(size_t)(rowbase + m) * NOUT + n] = acc[r] + bv;
    }
  }
}

// ---------------------------------------------------------------------------
// Fused MLP branch kernel.  MODE: 0=encoder 1=obs-decoder 2=action 3=transition
// One workgroup (8 wave32) per 16-row tile; hidden activations live in LDS,
// layers 2/3 are computed in-place (read-all -> barrier -> write-all).
// ---------------------------------------------------------------------------
template <int MODE>
__global__ __launch_bounds__(256) void mlp_fused(
    const float* __restrict__ inputx, const int* __restrict__ inputts,
    const float* __restrict__ con, const float* __restrict__ temb,
    const float* __restrict__ W1, const float* __restrict__ B1,
    const float* __restrict__ W2, const float* __restrict__ B2,
    const float* __restrict__ W3, const float* __restrict__ B3,
    const float* __restrict__ W4, const float* __restrict__ B4,
    float* __restrict__ outp) {
  constexpr int K1 = MODE == 0 ? 576 : MODE == 1 ? 87 : MODE == 2 ? 34 : 24;
  constexpr int NOUT = MODE == 0 ? 64 : MODE == 2 ? 6 : 18;
  constexpr int K1R = (K1 + 3) & ~3;
  constexpr int SX = K1R + 4;
  constexpr bool G1 = (K1R != K1);

  __shared__ float hbuf[16 * HS];
  __shared__ float xin[(MODE == 0) ? 16 : 16 * SX];

  const int tid = threadIdx.x;
  const int wave = tid >> 5, lane = tid & 31;
  const int half = lane >> 4, l15 = lane & 15;
  const int rowbase = blockIdx.x * 16;
  const int n0 = wave * 64;

  // ---- stage gathered layer-1 input into LDS (modes 1..3) ----
  if (MODE != 0) {
    for (int i = tid; i < 16 * SX; i += 256) {
      const int r = i / SX, j = i - r * SX;
      const int row = rowbase + r;
      float v = 0.f;
      if (j < K1) {
        if (MODE == 1) {  // [con(64) | t_emb(5) | xpt0(18)]
          const int b = row >> 5, hh = row & 31;
          if (j < 64) {
            v = con[b * 64 + j];
          } else if (j < 69) {
            int ti = inputts[b * 32 + hh] - inputts[b * 32] + 1;
            ti = ti < 0 ? 0 : (ti > 32 ? 32 : ti);
            v = temb[ti * EMBT + (j - 64)];
          } else {
            const int c = j - 69;
            v = inputx[(size_t)b * (HOR * TRANSD) + (c < OBS ? c : 23)];
          }
        } else if (MODE == 2) {  // [obs(t) | obs(t+1)]
          const int b = row / 31, hh = row - b * 31;
          if (j < OBS)
            v = inputx[(size_t)b * (HOR * TRANSD) + hh * TRANSD + j];
          else
            v = inputx[(size_t)b * (HOR * TRANSD) + (hh + 1) * TRANSD +
                       (j - OBS)];
        } else {  // MODE 3: raw transition rows
          v = inputx[(size_t)row * TRANSD + j];
        }
      }
      xin[i] = v;
    }
    __syncthreads();
  }

  // ---- layer 1 -> hbuf ----
  {
    v8f acc[4];
#pragma unroll
    for (int t = 0; t < 4; ++t)
#pragma unroll
      for (int r = 0; r < 8; ++r) acc[t][r] = 0.f;
    if (MODE == 0) {
      // A-fragments gathered straight from global (xpt concat, K=576)
      const int row = rowbase + l15;
      const float* xr = inputx + (size_t)row * (HOR * TRANSD);
      for (int k = 0; k < 576; k += 4) {
        const int ka = k + 2 * half;
        const int h0 = ka / 18, c0 = ka - h0 * 18;
        const int kb = ka + 1;
        const int h1 = kb / 18, c1 = kb - h1 * 18;
        v2f a;
        a.x = xr[h0 * TRANSD + (c0 < OBS ? c0 : 23)];
        a.y = xr[h1 * TRANSD + (c1 < OBS ? c1 : 23)];
#pragma unroll
        for (int t = 0; t < 4; ++t) {
          const int n = n0 + t * 16 + l15;
          v2f b = *(const v2f*)(W1 + (size_t)n * 576 + ka);
          acc[t] = wmma_f32(a, b, acc[t]);
        }
      }
    } else {
      layer_compute<K1, SX, G1>(W1, xin, half, l15, n0, acc);
    }
    layer_store(hbuf, B1, acc, half, l15, n0, true);
  }
  __syncthreads();

  // ---- layer 2 (in-place) ----
  {
    v8f acc[4];
#pragma unroll
    for (int t = 0; t < 4; ++t)
#pragma unroll
      for (int r = 0; r < 8; ++r) acc[t][r] = 0.f;
    layer_compute<HIDW, HS, false>(W2, hbuf, half, l15, n0, acc);
    __syncthreads();
    layer_store(hbuf, B2, acc, half, l15, n0, true);
  }
  __syncthreads();

  // ---- layer 3 (in-place) ----
  {
    v8f acc[4];
#pragma unroll
    for (int t = 0; t < 4; ++t)
#pragma unroll
      for (int r = 0; r < 8; ++r) acc[t][r] = 0.f;
    layer_compute<HIDW, HS, false>(W3, hbuf, half, l15, n0, acc);
    __syncthreads();
    layer_store(hbuf, B3, acc, half, l15, n0, true);
  }
  __syncthreads();

  // ---- output layer ----
  out_layer<NOUT>(W4, B4, hbuf, outp, rowbase, wave, half, l15);
}

// ---------------------------------------------------------------------------
// VQ: codebook norms, WMMA distance + argmin, post-processing, EMA scaling.
// ---------------------------------------------------------------------------
__global__ __launch_bounds__(256) void vq_init(float* __restrict__ nemb,
                                               float* __restrict__ cnt,
                                               unsigned long long* __restrict__ keys,
                                               float* __restrict__ loss) {
  const size_t i = (size_t)blockIdx.x * 256 + threadIdx.x;
  if (i < NEMB_ELEMS) nemb[i] = 0.f;
  if (i < NEMB) cnt[i] = 0.f;
  if (i < BATCH) keys[i] = ~0ull;
  if (i == 0) *loss = 0.f;
}

__global__ __launch_bounds__(256) void vq_norms(const float* __restrict__ emb,
                                                float* __restrict__ norms) {
  const int i = blockIdx.x * 256 + threadIdx.x;
  if (i < NEMB) {
    const float* e = emb + (size_t)i * CTX;
    float s = 0.f;
#pragma unroll
    for (int j = 0; j < CTX; ++j) s += e[j] * e[j];
    norms[i] = s;
  }
}

// grid: (64 row tiles, 31 chunks of 4096 codes); block 256 (8 waves x 512 codes)
__global__ __launch_bounds__(256) void vq_dist(
    const float* __restrict__ encx, const float* __restrict__ emb,
    const float* __restrict__ norms, unsigned long long* __restrict__ keys) {
  __shared__ float xs[16 * 68];
  const int tid = threadIdx.x;
  const int wave = tid >> 5, lane = tid & 31;
  const int half = lane >> 4, l15 = lane & 15;
  const int rowtile = blockIdx.x;

  for (int i = tid; i < 16 * 64; i += 256) {
    const int r = i >> 6, c = i & 63;
    xs[r * 68 + c] = encx[(rowtile * 16 + r) * CTX + c];
  }
  __syncthreads();

  v2f areg[16];
#pragma unroll
  for (int ks = 0; ks < 16; ++ks)
    areg[ks] = *(const v2f*)(xs + l15 * 68 + ks * 4 + 2 * half);

  const int cbase = blockIdx.y * 4096 + wave * 512;
  float bv[8];
  unsigned bi[8];
#pragma unroll
  for (int r = 0; r < 8; ++r) {
    bv[r] = __int_as_float(0x7F800000);  // +inf
    bi[r] = 0;
  }

  for (int t = 0; t < 32; ++t) {
    const int n = cbase + t * 16 + l15;
    const float* er = emb + (size_t)n * CTX;
    v8f acc;
#pragma unroll
    for (int r = 0; r < 8; ++r) acc[r] = 0.f;
#pragma unroll
    for (int ks = 0; ks < 16; ++ks) {
      v2f b = *(const v2f*)(er + ks * 4 + 2 * half);
      acc = wmma_f32(areg[ks], b, acc);
    }
    const float nn = norms[n];
#pragma unroll
    for (int r = 0; r < 8; ++r) {
      const float s = nn - 2.0f * acc[r];  // ||x||^2 omitted (row-constant)
      if (s < bv[r]) {
        bv[r] = s;
        bi[r] = (unsigned)n;
      }
    }
  }

#pragma unroll
  for (int r = 0; r < 8; ++r) {
    unsigned long long key = ((unsigned long long)fkey(bv[r]) << 32) | bi[r];
    for (int m = 1; m < 16; m <<= 1) {
      const unsigned long long o = shfl_xor_u64(key, m);
      if (o < key) key = o;
    }
    if (l15 == 0) {
      const int row = rowtile * 16 + r + 8 * half;
      atomicMin(&keys[row], key);
    }
  }
}

// 1024 blocks x 64 threads: con, loss, cluster counts, dw accumulation
__global__ __launch_bounds__(64) void vq_post(
    const unsigned long long* __restrict__ keys, const float* __restrict__ emb,
    const float* __restrict__ encx, float* __restrict__ con,
    float* __restrict__ nemb, float* __restrict__ cnt,
    float* __restrict__ loss) {
  const int b = blockIdx.x, j = threadIdx.x;
  const unsigned idx = (unsigned)(keys[b] & 0xFFFFFFFFu);
  const float x = encx[b * CTX + j];
  const float e = emb[(size_t)idx * CTX + j];
  con[b * CTX + j] = x + (e - x);  // straight-through estimator
  atomicAdd(&nemb[(size_t)idx * CTX + j], x);
  __shared__ float red[64];
  const float d = x - e;
  red[j] = d * d;
  __syncthreads();
  if (j == 0) {
    float s = 0.f;
    for (int i = 0; i < 64; ++i) s += red[i];
    atomicAdd(loss, s * (1.0f / 65536.0f));
    atomicAdd(&cnt[idx], 1.0f);
  }
}

__global__ __launch_bounds__(256) void vq_scale(float* __restrict__ nemb,
                                                const float* __restrict__ cnt) {
  const size_t i = (size_t)blockIdx.x * 256 + threadIdx.x;
  if (i < NEMB_ELEMS) {
    const float c = cnt[i >> 6];
    // ema_cluster = (c + eps) / (n + NEMB*eps) * n,  n = sum(counts) = 1024
    const float denom =
        (c + 1e-5f) * (1024.0f / (1024.0f + (float)NEMB * 1e-5f));
    nemb[i] = nemb[i] / denom;
  }
}

// ---------------------------------------------------------------------------
extern "C" void kernel_launch(void* const* d_in, const int* in_sizes, int n_in,
                              void* d_out, int out_size, void* d_ws,
                              size_t ws_size, hipStream_t stream) {
  (void)in_sizes;
  (void)n_in;
  (void)out_size;
  (void)ws_size;
  const float* inputx = (const float*)d_in[0];
  const int* inputts = (const int*)d_in[1];
  const float* enc1_w = (const float*)d_in[2];
  const float* enc1_b = (const float*)d_in[3];
  const float* enc2_w = (const float*)d_in[4];
  const float* enc2_b = (const float*)d_in[5];
  const float* enc3_w = (const float*)d_in[6];
  const float* enc3_b = (const float*)d_in[7];
  const float* enc4_w = (const float*)d_in[8];
  const float* enc4_b = (const float*)d_in[9];
  const float* dec_t_emb = (const float*)d_in[10];
  const float* do1_w = (const float*)d_in[11];
  const float* do1_b = (const float*)d_in[12];
  const float* do2_w = (const float*)d_in[13];
  const float* do2_b = (const float*)d_in[14];
  const float* do3_w = (const float*)d_in[15];
  const float* do3_b = (const float*)d_in[16];
  const float* do4_w = (const float*)d_in[17];
  const float* do4_b = (const float*)d_in[18];
  const float* a1_w = (const float*)d_in[19];
  const float* a1_b = (const float*)d_in[20];
  const float* a2_w = (const float*)d_in[21];
  const float* a2_b = (const float*)d_in[22];
  const float* a3_w = (const float*)d_in[23];
  const float* a3_b = (const float*)d_in[24];
  const float* a4_w = (const float*)d_in[25];
  const float* a4_b = (const float*)d_in[26];
  const float* t1_w = (const float*)d_in[27];
  const float* t1_b = (const float*)d_in[28];
  const float* t2_w = (const float*)d_in[29];
  const float* t2_b = (const float*)d_in[30];
  const float* t3_w = (const float*)d_in[31];
  const float* t3_b = (const float*)d_in[32];
  const float* t4_w = (const float*)d_in[33];
  const float* t4_b = (const float*)d_in[34];
  const float* embeddings = (const float*)d_in[35];

  float* out = (float*)d_out;
  float* con_o = out + OFF_CON;
  float* loss_o = out + OFF_LOSS;
  float* obs_o = out + OFF_OBS;
  float* act_o = out + OFF_ACT;
  float* tr_o = out + OFF_TRANSI;
  float* encx_o = out + OFF_ENCX;
  float* nemb_o = out + OFF_NEMB;

  // workspace: keys(8KB) | cnt(496KB) | norms(496KB)  (~1 MB total)
  unsigned long long* keys = (unsigned long long*)d_ws;
  float* cnt = (float*)((char*)d_ws + 8192);
  float* norms = cnt + NEMB;

  // 1) encoder -> enc_x
  mlp_fused<0><<<64, 256, 0, stream>>>(inputx, inputts, con_o, dec_t_emb,
                                       enc1_w, enc1_b, enc2_w, enc2_b, enc3_w,
                                       enc3_b, enc4_w, enc4_b, encx_o);
  // 2) VQ init / norms / distance-argmin / post / EMA scale
  vq_init<<<31744, 256, 0, stream>>>(nemb_o, cnt, keys, loss_o);
  vq_norms<<<496, 256, 0, stream>>>(embeddings, norms);
  vq_dist<<<dim3(64, 31), 256, 0, stream>>>(encx_o, embeddings, norms, keys);
  vq_post<<<1024, 64, 0, stream>>>(keys, embeddings, encx_o, con_o, nemb_o,
                                   cnt, loss_o);
  vq_scale<<<31744, 256, 0, stream>>>(nemb_o, cnt);
  // 3) decoders (obs needs con)
  mlp_fused<1><<<2048, 256, 0, stream>>>(inputx, inputts, con_o, dec_t_emb,
                                         do1_w, do1_b, do2_w, do2_b, do3_w,
                                         do3_b, do4_w, do4_b, obs_o);
  mlp_fused<2><<<1984, 256, 0, stream>>>(inputx, inputts, con_o, dec_t_emb,
                                         a1_w, a1_b, a2_w, a2_b, a3_w, a3_b,
                                         a4_w, a4_b, act_o);
  mlp_fused<3><<<2048, 256, 0, stream>>>(inputx, inputts, con_o, dec_t_emb,
                                         t1_w, t1_b, t2_w, t2_b, t3_w, t3_b,
                                         t4_w, t4_b, tr_o);
}